// GCNRegression_66743791780138
// MI455X (gfx1250) — compile-verified
//
#include <hip/hip_runtime.h>

typedef float v2f __attribute__((ext_vector_type(2)));
typedef float v8f __attribute__((ext_vector_type(8)));

#define NN      100000
#define NE      1600000
#define FIN     64
#define HID     128
#define NG      1024
#define EPSV    1e-5f

// ---------------- zero scratch ----------------
__global__ void zero_f4(float4* __restrict__ p, long n4) {
    long i = (long)blockIdx.x * blockDim.x + threadIdx.x;
    if (i < n4) p[i] = make_float4(0.f, 0.f, 0.f, 0.f);
}

// ---------------- edge scatter-add: agg[dst] += h[src] * ew ----------------
// one thread handles 4 consecutive feats; (d/4) consecutive threads = one edge
__global__ void scatter_edges(const float* __restrict__ h,
                              const int* __restrict__ src,
                              const int* __restrict__ dst,
                              const float* __restrict__ ew,
                              float* __restrict__ agg,
                              int d_shift /* log2(d/4) */) {
    long idx = (long)blockIdx.x * blockDim.x + threadIdx.x;
    long e = idx >> d_shift;
    if (e >= NE) return;
    int chunks = 1 << d_shift;
    int d = chunks << 2;
    int f = (int)(idx & (chunks - 1)) << 2;
    int s = src[e], t = dst[e];
    float w = ew[e];
    const float4 v = *(const float4*)(h + (long)s * d + f);
    float* a = agg + (long)t * d + f;
    atomicAdd(a + 0, v.x * w);
    atomicAdd(a + 1, v.y * w);
    atomicAdd(a + 2, v.z * w);
    atomicAdd(a + 3, v.w * w);
}

// ---------------- fused dual-GEMM + bias + BN + ReLU via f32 WMMA ----------------
// out[n,:] = relu(BN(agg[n,:]@Wr + br + h[n,:]@Wo))
// block = 256 thr = 8 waves; block -> 16 rows, wave w -> cols [16w,16w+16)
__global__ __launch_bounds__(256) void gemm_bn_relu(
        const float* __restrict__ agg, const float* __restrict__ h, int d,
        const float* __restrict__ wr, const float* __restrict__ br,
        const float* __restrict__ wo, const float* __restrict__ bg,
        const float* __restrict__ bb, const float* __restrict__ bm,
        const float* __restrict__ bv, float* __restrict__ out) {
    const int lane  = threadIdx.x & 31;
    const int wave  = threadIdx.x >> 5;
    const int mbase = blockIdx.x << 4;
    const int nbase = wave << 4;
    const int row   = mbase + (lane & 15);   // A-frag row (both lane halves)
    const int hi    = lane >> 4;             // 0: K=kb,kb+1  1: K=kb+2,kb+3
    const int col   = nbase + (lane & 15);   // B/C/D column

    v8f acc = {};
    // ---- K over agg x Wr ----
    for (int kb = 0; kb < d; kb += 4) {
        const int k = kb + 2 * hi;
        v2f a = *(const v2f*)(agg + (long)row * d + k);
        v2f b;
        b.x = wr[(long)k * HID + col];
        b.y = wr[(long)(k + 1) * HID + col];
        acc = __builtin_amdgcn_wmma_f32_16x16x4_f32(false, a, false, b,
                                                    (short)0, acc, false, false);
    }
    // ---- K over h x Wo (concatenated-K second half) ----
    for (int kb = 0; kb < d; kb += 4) {
        const int k = kb + 2 * hi;
        v2f a = *(const v2f*)(h + (long)row * d + k);
        v2f b;
        b.x = wo[(long)k * HID + col];
        b.y = wo[(long)(k + 1) * HID + col];
        acc = __builtin_amdgcn_wmma_f32_16x16x4_f32(false, a, false, b,
                                                    (short)0, acc, false, false);
    }
    // epilogue: col-dependent BN params (col fixed per lane)
    const float bias  = br[col] - bm[col];
    const float scale = bg[col] * rsqrtf(bv[col] + EPSV);
    const float beta  = bb[col];
#pragma unroll
    for (int j = 0; j < 8; ++j) {
        const int rr = mbase + j + 8 * hi;    // D-matrix row for VGPR j
        float val = (acc[j] + bias) * scale + beta;
        out[(long)rr * HID + col] = val > 0.f ? val : 0.f;
    }
}

// ---------------- mean-pool accumulation ----------------
// 32 threads per node (4 feats each); lane f==0 also bumps the count
__global__ void pool_kernel(const float* __restrict__ h,
                            const int* __restrict__ batch,
                            float* __restrict__ sums,
                            float* __restrict__ cnt) {
    long idx = (long)blockIdx.x * blockDim.x + threadIdx.x;
    long n = idx >> 5;
    if (n >= NN) return;
    int f = (int)(idx & 31) << 2;
    int g = batch[n];
    const float4 v = *(const float4*)(h + n * HID + f);
    float* s = sums + (long)g * HID + f;
    atomicAdd(s + 0, v.x);
    atomicAdd(s + 1, v.y);
    atomicAdd(s + 2, v.z);
    atomicAdd(s + 3, v.w);
    if (f == 0) atomicAdd(&cnt[g], 1.0f);
}

// ---------------- MLP head: relu(pooled@W1+b1)@W2+b2 ----------------
__global__ void head_kernel(const float* __restrict__ sums,
                            const float* __restrict__ cnt,
                            const float* __restrict__ w1, const float* __restrict__ b1,
                            const float* __restrict__ w2, const float* __restrict__ b2,
                            float* __restrict__ out) {
    __shared__ float part[64];
    const int g = blockIdx.x;
    const int t = threadIdx.x;          // 64 threads
    const float inv = 1.0f / fmaxf(cnt[g], 1.0f);
    float acc = b1[t];
    for (int k = 0; k < HID; ++k)
        acc += sums[(long)g * HID + k] * inv * w1[k * 64 + t];
    acc = fmaxf(acc, 0.f);
    part[t] = acc * w2[t];
    __syncthreads();
    for (int s = 32; s > 0; s >>= 1) {
        if (t < s) part[t] += part[t + s];
        __syncthreads();
    }
    if (t == 0) out[g] = part[0] + b2[0];
}

extern "C" void kernel_launch(void* const* d_in, const int* in_sizes, int n_in,
                              void* d_out, int out_size, void* d_ws, size_t ws_size,
                              hipStream_t stream) {
    const float* x     = (const float*)d_in[0];
    const int*   ei    = (const int*)d_in[1];
    const float* ea    = (const float*)d_in[2];
    const int*   batch = (const int*)d_in[3];
    const int*   src   = ei;
    const int*   dst   = ei + NE;

    char* ws = (char*)d_ws;
    const size_t HBYTES = (size_t)NN * HID * sizeof(float);   // 51.2 MB
    float* agg  = (float*)(ws);
    float* hA   = (float*)(ws + HBYTES);
    float* hB   = (float*)(ws + 2 * HBYTES);
    float* pool = (float*)(ws + 3 * HBYTES);                  // NG*HID floats
    float* cnt  = pool + (size_t)NG * HID;                    // NG floats
    float* out  = (float*)d_out;

    // zero pool + counts (contiguous, 132096 floats = 33024 float4)
    {
        long n4 = ((long)NG * HID + NG) / 4;
        zero_f4<<<(int)((n4 + 255) / 256), 256, 0, stream>>>((float4*)pool, n4);
    }

    const float* hcur = x;
    float* dests[3] = { hA, hB, hA };
    for (int L = 0; L < 3; ++L) {
        const int d = (L == 0) ? FIN : HID;
        const float* wr  = (const float*)d_in[4 + 7 * L + 0];
        const float* br  = (const float*)d_in[4 + 7 * L + 1];
        const float* wo  = (const float*)d_in[4 + 7 * L + 2];
        const float* bg  = (const float*)d_in[4 + 7 * L + 3];
        const float* bbt = (const float*)d_in[4 + 7 * L + 4];
        const float* bm  = (const float*)d_in[4 + 7 * L + 5];
        const float* bv  = (const float*)d_in[4 + 7 * L + 6];

        long aggN4 = (long)NN * d / 4;
        zero_f4<<<(int)((aggN4 + 255) / 256), 256, 0, stream>>>((float4*)agg, aggN4);

        const int shift = (d == 64) ? 4 : 5;        // log2(d/4)
        long nth = (long)NE << shift;
        scatter_edges<<<(int)((nth + 255) / 256), 256, 0, stream>>>(
            hcur, src, dst, ea, agg, shift);

        gemm_bn_relu<<<NN / 16, 256, 0, stream>>>(
            agg, hcur, d, wr, br, wo, bg, bbt, bm, bv, dests[L]);
        hcur = dests[L];
    }

    long pn = (long)NN * 32;
    pool_kernel<<<(int)((pn + 255) / 256), 256, 0, stream>>>(hcur, batch, pool, cnt);

    head_kernel<<<NG, 64, 0, stream>>>(pool, cnt,
                                       (const float*)d_in[25], (const float*)d_in[26],
                                       (const float*)d_in[27], (const float*)d_in[28],
                                       out);
}